// BaseModule_71863392797203
// MI455X (gfx1250) — compile-verified
//
#include <hip/hip_runtime.h>
#include <hip/hip_bf16.h>

// ---------------- problem constants (match reference setup_inputs) ----------
#define Bb 2
#define Vv 2
#define HC 512
#define WC 512
#define Hf 256
#define Wf 256
#define Cc 64
#define HW (Hf * Wf)          // 65536 points / pixels per view
#define BV (Bb * Vv)          // 4 views total
#define N_PTS HW              // sentinel for "no winner"
#define EPSF 1.0e-4f
#define BIGF 1.0e10f

// output layout (floats): prj_fs (B,V,C,H,W) ++ warped (V,B,3,H,W) ++ prj_depths (B,V,1,H,W)
#define OFF_FS   ((size_t)0)
#define OFF_WARP ((size_t)BV * Cc * HW)                 // 16,777,216
#define OFF_DEP  (OFF_WARP + (size_t)BV * 3 * HW)       // 17,563,648

// workspace layout (bytes, 256-aligned)
#define WS_MATS      ((size_t)0)                         // BV*16 floats
#define WS_COLORS_R  ((size_t)256)                       // BV*3*HW floats
#define WS_DEPTHS_R  (WS_COLORS_R + (size_t)BV * 3 * HW * 4)
#define WS_ZBUF      (WS_DEPTHS_R + (size_t)BV * HW * 4)
#define WS_WIN       (WS_ZBUF     + (size_t)BV * HW * 4)
#define WS_PIDX      (WS_WIN      + (size_t)BV * HW * 4)
#define WS_PZ        (WS_PIDX     + (size_t)BV * HW * 4)

typedef float v2f __attribute__((ext_vector_type(2)));
typedef float v8f __attribute__((ext_vector_type(8)));

// ---------------- tiny 4x4 helpers --------------------------------------
__device__ inline void mat4_mul(const float* A, const float* Bm, float* Cm) {
  for (int r = 0; r < 4; ++r)
    for (int c = 0; c < 4; ++c) {
      float s = 0.f;
      for (int k = 0; k < 4; ++k) s += A[r * 4 + k] * Bm[k * 4 + c];
      Cm[r * 4 + c] = s;
    }
}

__device__ inline void mat4_inv(const float* A, float* inv) {
  float m[4][8];
  for (int r = 0; r < 4; ++r)
    for (int c = 0; c < 4; ++c) { m[r][c] = A[r * 4 + c]; m[r][4 + c] = (r == c) ? 1.f : 0.f; }
  for (int col = 0; col < 4; ++col) {
    int piv = col; float best = fabsf(m[col][col]);
    for (int r = col + 1; r < 4; ++r) { float v = fabsf(m[r][col]); if (v > best) { best = v; piv = r; } }
    if (piv != col)
      for (int c = 0; c < 8; ++c) { float t = m[col][c]; m[col][c] = m[piv][c]; m[piv][c] = t; }
    float rinv = 1.0f / m[col][col];
    for (int c = 0; c < 8; ++c) m[col][c] *= rinv;
    for (int r = 0; r < 4; ++r) {
      if (r == col) continue;
      float f = m[r][col];
      for (int c = 0; c < 8; ++c) m[r][c] -= f * m[col][c];
    }
  }
  for (int r = 0; r < 4; ++r)
    for (int c = 0; c < 4; ++c) inv[r * 4 + c] = m[r][4 + c];
}

// Mat_{b,v} = sK_b * dstRT_b * srcRTinv_{b,v} * inv(sK_b)
__global__ void mats_kernel(const float* __restrict__ K,
                            const float* __restrict__ srcRTinv,
                            const float* __restrict__ dstRT,
                            float* __restrict__ Mats) {
  int i = threadIdx.x;
  if (i >= BV) return;
  int b = i / Vv, v = i % Vv;
  float sK[16], sKinv[16], t0[16], t1[16];
  const float scl[4] = {0.5f, 0.5f, 1.f, 1.f};  // sw=Wf/Wc, sh=Hf/Hc scale rows 0,1
  for (int r = 0; r < 4; ++r)
    for (int c = 0; c < 4; ++c) sK[r * 4 + c] = K[b * 16 + r * 4 + c] * scl[r];
  mat4_inv(sK, sKinv);
  const float* dst  = dstRT + (size_t)b * 16;                 // dst_RTs[:,0]
  const float* sinv = srcRTinv + ((size_t)b * Vv + v) * 16;
  mat4_mul(dst, sinv, t0);
  mat4_mul(sK, t0, t1);
  mat4_mul(t1, sKinv, t0);
  for (int k = 0; k < 16; ++k) Mats[i * 16 + k] = t0[k];
}

__global__ void init_kernel(float* __restrict__ zbuf, int* __restrict__ win) {
  int i = blockIdx.x * blockDim.x + threadIdx.x;
  if (i < BV * HW) { zbuf[i] = BIGF; win[i] = N_PTS; }
}

// nearest depth (index 2i+1 per dim) + antialiased linear colors (raw taps 1/4,3/4,3/4,1/4)
__global__ void resize_kernel(const float* __restrict__ depths,
                              const float* __restrict__ colors,
                              float* __restrict__ depths_r,
                              float* __restrict__ colors_r) {
  int i = blockIdx.x * blockDim.x + threadIdx.x;
  if (i >= BV * HW) return;
  int view = i / HW;
  int pix  = i - view * HW;
  int y = pix >> 8, x = pix & 255;

  depths_r[i] = depths[(size_t)view * (HC * WC) + (size_t)(2 * y + 1) * WC + (2 * x + 1)];

  const float wraw[4] = {0.25f, 0.75f, 0.75f, 0.25f};
  float wy[4], wx[4]; float sy = 0.f, sx = 0.f;
  int jy[4], jx[4];
  for (int t = 0; t < 4; ++t) {
    int j = 2 * y - 1 + t; jy[t] = j;
    if (j >= 0 && j < HC) { wy[t] = wraw[t]; sy += wraw[t]; } else wy[t] = 0.f;
    int k = 2 * x - 1 + t; jx[t] = k;
    if (k >= 0 && k < WC) { wx[t] = wraw[t]; sx += wraw[t]; } else wx[t] = 0.f;
  }
  float norm = 1.0f / (sy * sx);
  for (int ch = 0; ch < 3; ++ch) {
    const float* src = colors + ((size_t)view * 3 + ch) * (HC * WC);
    float acc = 0.f;
    for (int t = 0; t < 4; ++t) {
      if (wy[t] == 0.f) continue;
      float rowacc = 0.f;
      for (int s = 0; s < 4; ++s) {
        if (wx[s] == 0.f) continue;
        rowacc += wx[s] * src[(size_t)jy[t] * WC + jx[s]];
      }
      acc += wy[t] * rowacc;
    }
    colors_r[((size_t)view * 3 + ch) * HW + pix] = acc * norm;
  }
}

// WMMA point transform + z scatter-min.
// A(16x4) = Mat rows 0..3 (rest zero), B(4x16) = 16 points' [gx*d, gy*d, d, 1].
// ISA layouts: A lane<16 -> (M=ln, K=0,1), lane>=16 -> (M=ln, K=2,3).
//              B lane<16 -> (N=ln, K=0,1), lane>=16 -> (N=ln, K=2,3).
//              D VGPR r, lane n<16 -> element (M=r, N=n)  => q0,q1,q2 in acc[0..2] of lanes 0..15.
__global__ void splat_z_kernel(const float* __restrict__ depths_r,
                               const float* __restrict__ Mats,
                               float* __restrict__ zbuf,
                               int* __restrict__ pidx,
                               float* __restrict__ pz) {
  const int view = blockIdx.y;
  const int lane = threadIdx.x & 31;
  const int wave = threadIdx.x >> 5;
  const int ln   = lane & 15;
  const int hi   = lane >> 4;
  const int pt   = (blockIdx.x * (blockDim.x >> 5) + wave) * 16 + ln;

  const float* M = Mats + view * 16;
  v2f a;
  a.x = (ln < 4) ? M[ln * 4 + 2 * hi + 0] : 0.0f;
  a.y = (ln < 4) ? M[ln * 4 + 2 * hi + 1] : 0.0f;

  const int py = pt >> 8;
  const int px = pt & 255;
  const float gx = (float)px * (2.0f / 255.0f) - 1.0f;
  const float gy = (float)py * (2.0f / 255.0f) - 1.0f;
  const float d  = depths_r[(size_t)view * HW + pt];
  v2f bv;
  bv.x = hi ? d    : gx * d;
  bv.y = hi ? 1.0f : gy * d;

  v8f acc = {0.f, 0.f, 0.f, 0.f, 0.f, 0.f, 0.f, 0.f};
  acc = __builtin_amdgcn_wmma_f32_16x16x4_f32(false, a, false, bv, (short)0, acc,
                                              false, false);

  if (hi == 0) {
    float q0 = acc[0], q1 = acc[1], z = acc[2];
    bool small = fabsf(z) < EPSF;
    float zs = small ? EPSF : z;
    float sxv = small ? -10.0f : q0 / zs;
    float syv = small ? -10.0f : q1 / zs;
    float szv = small ? -10.0f : z;
    float fx = rintf((sxv + 1.0f) * 0.5f * (float)(Wf - 1));  // round half-to-even
    float fy = rintf((syv + 1.0f) * 0.5f * (float)(Hf - 1));
    bool valid = (fx >= 0.f) && (fx < (float)Wf) && (fy >= 0.f) && (fy < (float)Hf) &&
                 (szv > EPSF);
    int ipx = (int)fminf(fmaxf(fx, -1.0f), (float)Wf);
    int ipy = (int)fminf(fmaxf(fy, -1.0f), (float)Hf);
    int idx = valid ? (ipy * Wf + ipx) : -1;
    pidx[(size_t)view * HW + pt] = idx;
    pz[(size_t)view * HW + pt]   = szv;
    if (valid) {
      // positive floats: int-bit ordering == float ordering
      atomicMin((int*)(zbuf + (size_t)view * HW + idx), __float_as_int(szv));
    }
  }
}

__global__ void splat_win_kernel(const int* __restrict__ pidx,
                                 const float* __restrict__ pz,
                                 const float* __restrict__ zbuf,
                                 int* __restrict__ win) {
  int i = blockIdx.x * blockDim.x + threadIdx.x;
  if (i >= BV * HW) return;
  int view = i / HW;
  int pt   = i - view * HW;
  int idx  = pidx[i];
  if (idx >= 0) {
    float z = pz[i];
    if (z <= zbuf[(size_t)view * HW + idx]) atomicMin(&win[view * HW + idx], pt);
  }
}

__global__ void gather_kernel(const float* __restrict__ feats,
                              const float* __restrict__ colors_r,
                              const float* __restrict__ zbuf,
                              const int* __restrict__ win,
                              float* __restrict__ out) {
  int i = blockIdx.x * blockDim.x + threadIdx.x;
  if (i >= BV * HW) return;
  int view = i / HW;
  int pix  = i - view * HW;
  int wn = win[i];
  bool has = wn < N_PTS;
  int g = has ? wn : 0;

  // prj_fs (B,V,C,H,W): view-major == (b*V+v)
  const float* fsrc = feats + (size_t)view * Cc * HW;
  float* fdst = out + OFF_FS + (size_t)view * Cc * HW;
#pragma unroll 8
  for (int c = 0; c < Cc; ++c)
    fdst[(size_t)c * HW + pix] = has ? fsrc[(size_t)c * HW + g] : 0.0f;

  // warped (V,B,3,H,W)  (reference does NOT transpose this one)
  int b = view / Vv, v = view - b * Vv;
  for (int ch = 0; ch < 3; ++ch) {
    float val = has ? (colors_r[((size_t)view * 3 + ch) * HW + g] * 0.5f + 0.5f) : 0.0f;
    out[OFF_WARP + ((size_t)(v * Bb + b) * 3 + ch) * HW + pix] = val;
  }

  // prj_depths (B,V,1,H,W)
  float zb = zbuf[i];
  out[OFF_DEP + (size_t)view * HW + pix] = (zb < BIGF) ? zb : 0.0f;
}

extern "C" void kernel_launch(void* const* d_in, const int* in_sizes, int n_in,
                              void* d_out, int out_size, void* d_ws, size_t ws_size,
                              hipStream_t stream) {
  const float* depths   = (const float*)d_in[0];
  const float* colors   = (const float*)d_in[1];
  const float* feats    = (const float*)d_in[2];
  const float* Km       = (const float*)d_in[3];
  const float* srcRTinv = (const float*)d_in[5];
  const float* dstRT    = (const float*)d_in[6];
  float* out = (float*)d_out;

  char* ws = (char*)d_ws;
  float* mats     = (float*)(ws + WS_MATS);
  float* colors_r = (float*)(ws + WS_COLORS_R);
  float* depths_r = (float*)(ws + WS_DEPTHS_R);
  float* zbuf     = (float*)(ws + WS_ZBUF);
  int*   win      = (int*)(ws + WS_WIN);
  int*   pidx     = (int*)(ws + WS_PIDX);
  float* pz       = (float*)(ws + WS_PZ);

  mats_kernel<<<1, 32, 0, stream>>>(Km, srcRTinv, dstRT, mats);

  const int n = BV * HW;
  const int blocks = (n + 255) / 256;
  init_kernel<<<blocks, 256, 0, stream>>>(zbuf, win);
  resize_kernel<<<blocks, 256, 0, stream>>>(depths, colors, depths_r, colors_r);

  dim3 g(HW / 128, BV);  // 256 thr = 8 waves; each wave does one 16-point WMMA
  splat_z_kernel<<<g, 256, 0, stream>>>(depths_r, mats, zbuf, pidx, pz);
  splat_win_kernel<<<blocks, 256, 0, stream>>>(pidx, pz, zbuf, win);
  gather_kernel<<<blocks, 256, 0, stream>>>(feats, colors_r, zbuf, win, out);
}